// MPSLayer_28501402977028
// MI455X (gfx1250) — compile-verified
//
#include <hip/hip_runtime.h>

// MPSLayer on MI455X (gfx1250).
// out[b,:] = (prod_i x[b,i]) * (ones(32) @ prod_i C_i @ proj) + bias
// Memory-bound formulation (~29MB ~ 1.3us @ 23.3 TB/s). The 32x32 core-matrix
// chain is tree-parallelized (64x16 -> 8x8 -> 1x8, depth 29 matmuls) and each
// 32x32x32 step uses V_WMMA_F32_16X16X4_F32 (wave32, one wave/WG => EXEC all 1s).
// Segment matrices are prefetched to LDS up front via GLOBAL_LOAD_ASYNC_TO_LDS_B128
// (ASYNCcnt), so chain steps never touch HBM.

#define BATCH   4096
#define IN_DIM  1024
#define BOND    32
#define OUT_DIM 512
#define MATF    (BOND * BOND)   // 1024 floats per 32x32 matrix

typedef float v2f __attribute__((ext_vector_type(2)));
typedef float v8f __attribute__((ext_vector_type(8)));
typedef int   v4i __attribute__((vector_size(16)));   // matches builtin param type

#if defined(__has_builtin)
#if __has_builtin(__builtin_amdgcn_global_load_async_to_lds_b128)
#define HAVE_ASYNC_LDS 1
#endif
#endif

// Copy n_floats (multiple of 128) from global to LDS with one wave.
__device__ __forceinline__ void copy_seg_to_lds(float* lds_dst, const float* gsrc,
                                                int n_floats, int lane) {
#ifdef HAVE_ASYNC_LDS
  typedef __attribute__((address_space(1))) char gchar;
  typedef __attribute__((address_space(3))) char lchar;
  gchar* g = (gchar*)gsrc;
  lchar* l = (lchar*)lds_dst;
  const int bytes = n_floats * 4;
  for (int off = lane * 16; off < bytes; off += 32 * 16) {
    __builtin_amdgcn_global_load_async_to_lds_b128(
        (__attribute__((address_space(1))) v4i*)(g + off),
        (__attribute__((address_space(3))) v4i*)(l + off), 0, 0);
  }
  asm volatile("s_wait_asynccnt 0x0" ::: "memory");
#else
  const float4* g4 = (const float4*)gsrc;
  float4* l4 = (float4*)lds_dst;
  for (int i = lane; i < n_floats / 4; i += 32) l4[i] = g4[i];
#endif
  __syncthreads();
}

// P (32x32 row-major, LDS) <- P @ C (32x32 row-major, LDS). Single wave32.
// A layout (16x4 f32): lane holds row M=lane%16; (chunk c, half h, word w) = A[M][4c+2h+w]
// B layout (4x16 f32): lane holds col N=lane%16; (chunk c, half h, word w) = B[4c+2h+w][N]
// D layout (16x16 f32): VGPR k holds row M = k + 8*(lane/16), col N = lane%16
__device__ __forceinline__ void mm32_step(float* P, const float* C, int lane) {
  const int l16 = lane & 15;
  const int hf  = (lane >> 4) & 1;

  v2f a[2][8];
  v2f b[2][8];
#pragma unroll
  for (int mt = 0; mt < 2; ++mt) {
#pragma unroll
    for (int c = 0; c < 8; ++c) {
      const int row = mt * 16 + l16;
      const int k0  = 4 * c + 2 * hf;
      a[mt][c].x = P[row * 32 + k0];
      a[mt][c].y = P[row * 32 + k0 + 1];
    }
  }
#pragma unroll
  for (int nt = 0; nt < 2; ++nt) {
#pragma unroll
    for (int c = 0; c < 8; ++c) {
      const int col = nt * 16 + l16;
      const int k0  = 4 * c + 2 * hf;
      b[nt][c].x = C[(k0)     * 32 + col];
      b[nt][c].y = C[(k0 + 1) * 32 + col];
    }
  }

  v8f acc[2][2];
#pragma unroll
  for (int mt = 0; mt < 2; ++mt) {
#pragma unroll
    for (int nt = 0; nt < 2; ++nt) {
      v8f z = {0.f, 0.f, 0.f, 0.f, 0.f, 0.f, 0.f, 0.f};
#pragma unroll
      for (int c = 0; c < 8; ++c) {
        z = __builtin_amdgcn_wmma_f32_16x16x4_f32(
            false, a[mt][c], false, b[nt][c], (short)0, z, false, false);
      }
      acc[mt][nt] = z;
    }
  }

  __syncthreads();  // single wave: cheap; all P reads precede P writes

#pragma unroll
  for (int mt = 0; mt < 2; ++mt) {
#pragma unroll
    for (int nt = 0; nt < 2; ++nt) {
#pragma unroll
      for (int k = 0; k < 8; ++k) {
        const int row = mt * 16 + k + 8 * hf;
        const int col = nt * 16 + l16;
        P[row * 32 + col] = acc[mt][nt][k];
      }
    }
  }
}

// One wave per block: product of SEG consecutive 32x32 matrices.
// Whole segment staged in LDS up front (SEG<=16 -> 64KB + 4KB of 320KB/WGP).
template <int SEG>
__global__ void __launch_bounds__(32)
chain_tree(const float* __restrict__ mats, float* __restrict__ partials) {
  __shared__ float Cb[SEG * MATF];
  __shared__ float P[MATF];
  const int lane = threadIdx.x;
  const float* base = mats + (size_t)blockIdx.x * SEG * MATF;

  copy_seg_to_lds(Cb, base, SEG * MATF, lane);

#pragma unroll 4
  for (int j = 0; j < 32; ++j) P[j * 32 + lane] = Cb[j * 32 + lane];
  __syncthreads();

  for (int t = 1; t < SEG; ++t) {
    mm32_step(P, Cb + t * MATF, lane);
    __syncthreads();
  }

  float* outp = partials + (size_t)blockIdx.x * MATF;
#pragma unroll 4
  for (int j = 0; j < 32; ++j) outp[j * 32 + lane] = P[j * 32 + lane];
}

// One wave: Q = prod of 8 partials; r = ones(32) @ Q; w = r @ projection.
__global__ void __launch_bounds__(32)
chain_final(const float* __restrict__ partials, const float* __restrict__ projection,
            float* __restrict__ w_out) {
  __shared__ float Cb[8 * MATF];
  __shared__ float P[MATF];
  __shared__ float r[BOND];
  const int lane = threadIdx.x;

  copy_seg_to_lds(Cb, partials, 8 * MATF, lane);

#pragma unroll 4
  for (int j = 0; j < 32; ++j) P[j * 32 + lane] = Cb[j * 32 + lane];
  __syncthreads();

  for (int t = 1; t < 8; ++t) {
    mm32_step(P, Cb + t * MATF, lane);
    __syncthreads();
  }

  float cs = 0.f;
#pragma unroll 4
  for (int j = 0; j < 32; ++j) cs += P[j * 32 + lane];
  r[lane] = cs;
  __syncthreads();

  for (int t = 0; t < OUT_DIM / 32; ++t) {
    const int o = t * 32 + lane;
    float acc = 0.f;
#pragma unroll 8
    for (int k = 0; k < 32; ++k) acc = fmaf(r[k], projection[k * OUT_DIM + o], acc);
    w_out[o] = acc;
  }
}

// One block per batch row: s = prod(x[b,:]) (block reduction), then
// out[b,o] = s*w[o] + bias[o] with float4 stores. Bulk of HBM traffic lives here.
__global__ void __launch_bounds__(128)
fused_out(const float* __restrict__ x, const float* __restrict__ w,
          const float* __restrict__ bias, float* __restrict__ out) {
  __shared__ float wp[4];
  const int b    = blockIdx.x;
  const int tid  = threadIdx.x;
  const int lane = tid & 31;
  const int wid  = tid >> 5;

  const float4* xr = (const float4*)(x + (size_t)b * IN_DIM);
  const float4 va = xr[tid];         // first 512 floats of the row
  const float4 vb = xr[tid + 128];   // last 512 floats
  float p = ((va.x * va.y) * (va.z * va.w)) * ((vb.x * vb.y) * (vb.z * vb.w));
#pragma unroll
  for (int off = 16; off > 0; off >>= 1) p *= __shfl_xor(p, off, 32);
  if (lane == 0) wp[wid] = p;
  __syncthreads();
  const float s = (wp[0] * wp[1]) * (wp[2] * wp[3]);

  const int o = tid * 4;
  const float4 wv = *(const float4*)(w + o);
  const float4 bv = *(const float4*)(bias + o);
  float4 rr;
  rr.x = fmaf(s, wv.x, bv.x);
  rr.y = fmaf(s, wv.y, bv.y);
  rr.z = fmaf(s, wv.z, bv.z);
  rr.w = fmaf(s, wv.w, bv.w);
  *(float4*)(out + (size_t)b * OUT_DIM + o) = rr;
}

extern "C" void kernel_launch(void* const* d_in, const int* in_sizes, int n_in,
                              void* d_out, int out_size, void* d_ws, size_t ws_size,
                              hipStream_t stream) {
  const float* inputs = (const float*)d_in[0];  // [4096,1024]
  const float* cores  = (const float*)d_in[1];  // [1024,32,32]
  const float* proj   = (const float*)d_in[2];  // [32,512]
  const float* bias   = (const float*)d_in[3];  // [512]
  float* out = (float*)d_out;                   // [4096,512]

  float* wsf = (float*)d_ws;
  float* P1 = wsf;                 // 64 * 1024 floats (256KB)
  float* P2 = wsf + 64 * MATF;     //  8 * 1024 floats (32KB)
  float* w  = wsf + 72 * MATF;     // 512 floats

  chain_tree<16><<<64, 32, 0, stream>>>(cores, P1);  // 1024 cores -> 64 partials
  chain_tree<8><<<8, 32, 0, stream>>>(P1, P2);       // 64 -> 8
  chain_final<<<1, 32, 0, stream>>>(P2, proj, w);    // 8 -> Q -> r -> w
  fused_out<<<BATCH, 128, 0, stream>>>(inputs, w, bias, out);
}